// InterpretableAttention_20126216750040
// MI455X (gfx1250) — compile-verified
//
#include <hip/hip_runtime.h>
#include <math.h>

#define DIM  512
#define SEQ  4096
#define BAT  8

typedef __attribute__((ext_vector_type(2))) float v2f;
typedef __attribute__((ext_vector_type(8))) float v8f;

// ---------------------------------------------------------------------------
// Kernel 1: per batch, q0 = x[b,0,:] @ Wq^T + bq ; v = q0 @ Wk
// (bk dropped: it only adds a per-batch constant to scores, cancels in softmax)
// ---------------------------------------------------------------------------
__global__ __launch_bounds__(256) void qv_kernel(const float* __restrict__ x,
                                                 const float* __restrict__ Wq,
                                                 const float* __restrict__ bq,
                                                 const float* __restrict__ Wk,
                                                 float* __restrict__ v_out)
{
    __shared__ float x0[DIM];
    __shared__ float q0[DIM];
    const int b = blockIdx.x;
    const int t = threadIdx.x;

    // load x[b,0,:]
    x0[t]       = x[(size_t)b * SEQ * DIM + t];
    x0[t + 256] = x[(size_t)b * SEQ * DIM + t + 256];
    __syncthreads();

    // q0[e] = dot(x0, Wq[e,:]) + bq[e]; one wave per row (coalesced 128B reads)
    const int wave = t >> 5, lane = t & 31;
    for (int i = 0; i < DIM / 8; ++i) {
        const int e = wave + (i << 3);
        const float* wrow = Wq + (size_t)e * DIM;
        float p = 0.f;
        #pragma unroll
        for (int j = 0; j < DIM / 32; ++j)
            p = fmaf(x0[j * 32 + lane], wrow[j * 32 + lane], p);
        #pragma unroll
        for (int off = 16; off > 0; off >>= 1)
            p += __shfl_xor(p, off, 32);
        if (lane == 0) q0[e] = p + bq[e];
    }
    __syncthreads();

    // v[d] = sum_e q0[e] * Wk[e,d]  (coalesced over d, LDS broadcast of q0[e])
    float a0 = 0.f, a1 = 0.f;
    for (int e = 0; e < DIM; ++e) {
        const float q = q0[e];
        a0 = fmaf(q, Wk[(size_t)e * DIM + t], a0);
        a1 = fmaf(q, Wk[(size_t)e * DIM + t + 256], a1);
    }
    v_out[b * DIM + t]       = a0;
    v_out[b * DIM + t + 256] = a1;
}

// ---------------------------------------------------------------------------
// Kernel 2: scores[row] = x[row,:] . v[b,:]  via V_WMMA_F32_16X16X4_F32.
// One wave = 16 rows. A = 16x4 f32 tile of x (ISA layout: lanes 0-15 K={0,1},
// lanes 16-31 K={2,3}); B[k][n] = v[k0+k] for all n, so every column of the
// 16x16 f32 accumulator carries the running dot product.
// ---------------------------------------------------------------------------
__global__ __launch_bounds__(256) void scores_kernel(const float* __restrict__ x,
                                                     const float* __restrict__ v_in,
                                                     float* __restrict__ scores)
{
    __shared__ __align__(16) float sv[DIM];
    const int t = threadIdx.x;
    const int block_row = blockIdx.x * 128;          // 8 waves * 16 rows
    const int b = block_row / SEQ;                   // uniform per block (128 | 4096)

    sv[t]       = v_in[b * DIM + t];
    sv[t + 256] = v_in[b * DIM + t + 256];
    __syncthreads();

    const int wave = t >> 5, lane = t & 31;
    const int rowbase = block_row + wave * 16;       // flat row over [BAT*SEQ]
    const int m  = lane & 15;
    const int hi = lane >> 4;                        // 0: K={0,1}, 1: K={2,3}
    const float* ap = x + (size_t)(rowbase + m) * DIM + 2 * hi;

    v8f c = {0.f, 0.f, 0.f, 0.f, 0.f, 0.f, 0.f, 0.f};

    #pragma unroll 4
    for (int k0 = 0; k0 < DIM; k0 += 4) {
        v2f a  = *(const v2f*)(ap);                  // x[row, k0+2*hi : +2]
        v2f bb = *(const v2f*)(&sv[k0 + 2 * hi]);    // v broadcast into all 16 cols
        c = __builtin_amdgcn_wmma_f32_16x16x4_f32(
                /*neg_a=*/false, a, /*neg_b=*/false, bb,
                /*c_mod=*/(short)0, c, /*reuse_a=*/false, /*reuse_b=*/false);
        ap += 4;
    }

    // All 16 columns identical: accumulator VGPR r holds score(rowbase+r) in
    // lanes 0-15 and score(rowbase+8+r) in lanes 16-31.
    if ((lane & 15) == 0) {
        const int roff = hi * 8;
        #pragma unroll
        for (int r = 0; r < 8; ++r)
            scores[rowbase + roff + r] = c[r];
    }
}

// ---------------------------------------------------------------------------
// Kernel 3: per-batch softmax over 4096 scores (max-subtracted, two-level
// reduce: wave shuffles + LDS across 8 waves).
// ---------------------------------------------------------------------------
__global__ __launch_bounds__(256) void softmax_kernel(const float* __restrict__ scores,
                                                      float* __restrict__ out)
{
    __shared__ float red[8];
    const int b = blockIdx.x, t = threadIdx.x;
    const int wave = t >> 5, lane = t & 31;
    const float* srow = scores + b * SEQ;

    float s[16];
    float m = -INFINITY;
    #pragma unroll
    for (int i = 0; i < 16; ++i) {
        s[i] = srow[t + 256 * i];
        m = fmaxf(m, s[i]);
    }
    #pragma unroll
    for (int off = 16; off > 0; off >>= 1)
        m = fmaxf(m, __shfl_xor(m, off, 32));
    if (lane == 0) red[wave] = m;
    __syncthreads();
    float bm = red[0];
    #pragma unroll
    for (int w = 1; w < 8; ++w) bm = fmaxf(bm, red[w]);

    float sum = 0.f;
    #pragma unroll
    for (int i = 0; i < 16; ++i) {
        s[i] = __expf(s[i] - bm);
        sum += s[i];
    }
    #pragma unroll
    for (int off = 16; off > 0; off >>= 1)
        sum += __shfl_xor(sum, off, 32);
    __syncthreads();
    if (lane == 0) red[wave] = sum;
    __syncthreads();
    float tot = 0.f;
    #pragma unroll
    for (int w = 0; w < 8; ++w) tot += red[w];
    const float inv = 1.f / tot;

    #pragma unroll
    for (int i = 0; i < 16; ++i)
        out[b * SEQ + t + 256 * i] = s[i] * inv;
}

// ---------------------------------------------------------------------------
extern "C" void kernel_launch(void* const* d_in, const int* in_sizes, int n_in,
                              void* d_out, int out_size, void* d_ws, size_t ws_size,
                              hipStream_t stream) {
    const float* x  = (const float*)d_in[0];
    const float* Wq = (const float*)d_in[1];
    const float* bq = (const float*)d_in[2];
    const float* Wk = (const float*)d_in[3];
    // d_in[4] (bk) intentionally unused: adds a per-batch constant to every
    // score, which the softmax normalization cancels exactly.

    float* v      = (float*)d_ws;                 // BAT*DIM floats   (16 KB)
    float* scores = (float*)d_ws + BAT * DIM;     // BAT*SEQ floats  (128 KB)

    qv_kernel<<<BAT, 256, 0, stream>>>(x, Wq, bq, Wk, v);
    scores_kernel<<<(BAT * SEQ) / 128, 256, 0, stream>>>(x, v, scores);
    softmax_kernel<<<BAT, 256, 0, stream>>>(scores, (float*)d_out);
}